// RecursiveNN_28123445854312
// MI455X (gfx1250) — compile-verified
//
#include <hip/hip_runtime.h>
#include <stdint.h>

// ---------------------------------------------------------------------------
// RecursiveNN root = sum of leaf embeddings:  out[b,d] = sum_l E[ids[b,l], d]
// B=4096 trees, L=64 leaves, D=300, VOCAB=100000, f32.
// Memory-bound gather+reduce -> CDNA5 Tensor Data Mover in GATHER mode
// (32-bit row indices, 8 rows per descriptor) DMAs embedding rows into LDS,
// double-buffered, one wave32 per tree. Consumption uses ds_load_b128 +
// packed f32 adds to keep the per-wave instruction count minimal.
// ---------------------------------------------------------------------------

typedef uint32_t tdm_v4u __attribute__((ext_vector_type(4)));
typedef int      tdm_v8i __attribute__((ext_vector_type(8)));
typedef int      tdm_v4i __attribute__((ext_vector_type(4)));
typedef float    f32x4   __attribute__((ext_vector_type(4)));

#define RECNN_DIM   300
#define RECNN_L     64
#define RECNN_VOCAB 100000
#define TDM_ROWS    8                      // rows per gather descriptor (32-bit idx)
#define TDM_GROUPS  (RECNN_L / TDM_ROWS)   // 8 descriptors per tree
#define BUF_ELEMS   (TDM_ROWS * RECNN_DIM) // 2400 floats = 9600 B per buffer
#define ROW_QUADS   (RECNN_DIM / 4)        // 75 float4 per row
#define TAIL_LANES  (ROW_QUADS - 64)       // 11 lanes carry the 3rd quad

// Issue one TDM gather of 8 embedding rows (indices wid[base..base+7]) to LDS.
__device__ __forceinline__ void tdm_gather8(const int* __restrict__ wid, int base,
                                            uint32_t lds_addr,
                                            uint32_t ga_lo, uint32_t ga_hi,
                                            tdm_v8i g1)
{
    // D# group 0: count=1 (valid), gather_index_size=1 (32-bit), gather_mode=1,
    //             lds_addr, global_addr[56:0], type=2 ("image").
    tdm_v4u g0;
    g0.x = 1u | (1u << 30) | (1u << 31);
    g0.y = lds_addr;
    g0.z = ga_lo;
    g0.w = (ga_hi & 0x01FFFFFFu) | (2u << 30);

    // Groups 2/3 in gather mode: row_index_0..7 (32-bit mode). ids are
    // wave-uniform (one tree per block) -> readfirstlane; the compiler
    // folds these into s_load (SMEM) since the address is uniform.
    tdm_v4i i01, i23;
    i01.x = __builtin_amdgcn_readfirstlane(wid[base + 0]);
    i01.y = __builtin_amdgcn_readfirstlane(wid[base + 1]);
    i01.z = __builtin_amdgcn_readfirstlane(wid[base + 2]);
    i01.w = __builtin_amdgcn_readfirstlane(wid[base + 3]);
    i23.x = __builtin_amdgcn_readfirstlane(wid[base + 4]);
    i23.y = __builtin_amdgcn_readfirstlane(wid[base + 5]);
    i23.z = __builtin_amdgcn_readfirstlane(wid[base + 6]);
    i23.w = __builtin_amdgcn_readfirstlane(wid[base + 7]);

    tdm_v8i gx;
    gx[0] = 0; gx[1] = 0; gx[2] = 0; gx[3] = 0;
    gx[4] = 0; gx[5] = 0; gx[6] = 0; gx[7] = 0;

    __builtin_amdgcn_tensor_load_to_lds(g0, g1, i01, i23, gx, /*cpol=*/0);
}

__global__ __launch_bounds__(32)
void recnn_gather_sum(const int* __restrict__ word_ids,
                      const float* __restrict__ embedding,
                      float* __restrict__ out)
{
    // Double buffer: 2 x 8 rows x 300 f32 (rows contiguous, 1200 B each).
    __shared__ __align__(16) float buf[2 * BUF_ELEMS];

    const int  lane = threadIdx.x;
    const int  b    = blockIdx.x;
    const int* wid  = word_ids + (size_t)b * RECNN_L;

    const uint64_t ebase = (uint64_t)(uintptr_t)embedding;
    const uint32_t ga_lo = (uint32_t)ebase;
    const uint32_t ga_hi = (uint32_t)(ebase >> 32);

    const uint32_t lds0 = (uint32_t)(uintptr_t)(&buf[0]);
    const uint32_t lds1 = (uint32_t)(uintptr_t)(&buf[BUF_ELEMS]);

    // D# group 1 (compile-time constants):
    //  [17:16]  data_size=2 (4 bytes)
    //  [79:48]  tensor_dim0 = 300, [111:80] tensor_dim1 = 100000
    //  [127:112]tile_dim0 = 300, [143:128] tile_dim1 = 8 (# gather indices)
    //  [207:160]tensor_dim0_stride = 300
    tdm_v8i g1;
    g1[0] = (int)(2u << 16);
    g1[1] = (int)(((uint32_t)RECNN_DIM & 0xFFFFu) << 16);
    g1[2] = (int)((((uint32_t)RECNN_DIM >> 16) & 0xFFFFu) |
                  (((uint32_t)RECNN_VOCAB & 0xFFFFu) << 16));
    g1[3] = (int)((((uint32_t)RECNN_VOCAB >> 16) & 0xFFFFu) |
                  (((uint32_t)RECNN_DIM) << 16));
    g1[4] = TDM_ROWS;
    g1[5] = RECNN_DIM;
    g1[6] = 0;
    g1[7] = 0;

    // Per-lane accumulators: row quad indices lane, lane+32, lane+64.
    f32x4 a0 = {0.f, 0.f, 0.f, 0.f};
    f32x4 a1 = {0.f, 0.f, 0.f, 0.f};
    f32x4 a2 = {0.f, 0.f, 0.f, 0.f};   // valid only for lane < TAIL_LANES

    // Prime the double-buffered pipeline.
    tdm_gather8(wid, 0 * TDM_ROWS, lds0, ga_lo, ga_hi, g1);
    tdm_gather8(wid, 1 * TDM_ROWS, lds1, ga_lo, ga_hi, g1);

#pragma unroll
    for (int g = 0; g < TDM_GROUPS; ++g) {
        // Tensor ops from one wave complete in order: after issuing g+2 total,
        // tensorcnt<=1 guarantees group g has fully landed in LDS.
        if (g < TDM_GROUPS - 1) __builtin_amdgcn_s_wait_tensorcnt(1);
        else                    __builtin_amdgcn_s_wait_tensorcnt(0);

        const float* bufp = &buf[(g & 1) ? BUF_ELEMS : 0];
#pragma unroll
        for (int r = 0; r < TDM_ROWS; ++r) {
            // ds_load_b128, conflict-free (consecutive lanes -> consecutive 16B).
            const f32x4* row4 = (const f32x4*)(bufp + r * RECNN_DIM);
            a0 += row4[lane];
            a1 += row4[lane + 32];
            // Unconditional: lanes >= TAIL_LANES read past the row (later rows
            // or out-of-range LDS, which returns zero) -- garbage lands in an
            // accumulator whose store is masked once at the end. This avoids
            // toggling EXEC on every one of the 64 rows.
            a2 += row4[lane + 64];
        }

        if (g + 2 < TDM_GROUPS) {
            // Ensure our LDS reads of this buffer retired before the TDM may
            // overwrite it; also stops the compiler sinking loads past issue.
            asm volatile("s_wait_dscnt 0" ::: "memory");
            tdm_gather8(wid, (g + 2) * TDM_ROWS, (g & 1) ? lds1 : lds0,
                        ga_lo, ga_hi, g1);
        }
    }

    // Coalesced global_store_b128 of the 75 output quads per tree.
    f32x4* ob4 = (f32x4*)(out + (size_t)b * RECNN_DIM);
    ob4[lane]      = a0;
    ob4[lane + 32] = a1;
    if (lane < TAIL_LANES) ob4[lane + 64] = a2;
}

extern "C" void kernel_launch(void* const* d_in, const int* in_sizes, int n_in,
                              void* d_out, int out_size, void* d_ws, size_t ws_size,
                              hipStream_t stream) {
    (void)n_in; (void)out_size; (void)d_ws; (void)ws_size;
    const int*   word_ids  = (const int*)d_in[0];    // [B, 64] int32
    const float* embedding = (const float*)d_in[1];  // [100000, 300] f32
    float*       out       = (float*)d_out;          // [B, 300] f32

    const int B = in_sizes[0] / RECNN_L;             // 4096
    recnn_gather_sum<<<B, 32, 0, stream>>>(word_ids, embedding, out);
}